// GPModel_72825465471614
// MI455X (gfx1250) — compile-verified
//
#include <hip/hip_runtime.h>
#include <math.h>

// ---------------- problem constants (from reference setup_inputs) ----------
#define N_    2048
#define BS_   64
#define NB_   (N_/BS_)      // 32
#define DIN_  10
#define DP_   12            // padded feature dim for WMMA K=12
#define E_    6
#define D_    6

#if defined(__has_builtin)
#  if __has_builtin(__builtin_amdgcn_wmma_f32_16x16x4_f32)
#    define HAVE_WMMA_F32 1
#  endif
#endif
#ifndef HAVE_WMMA_F32
#  define HAVE_WMMA_F32 0
#endif

typedef float v2f __attribute__((ext_vector_type(2)));
typedef float v8f __attribute__((ext_vector_type(8)));

// ===========================================================================
// Generic fp32 WMMA GEMM:  Cdst(16x16 tile) = [Csrc] + scale * opA(A) @ opB(B)
// one wave (32 threads) per 16x16 output tile; K must be a multiple of 64.
// transA: A stored K-major (sA[m][k] = A[k*lda+m]);  transB: B stored N-major.
// VGPR fragment layout per CDNA5 ISA 7.12.2:
//   A 16x4 f32: lanes0-15 m=0..15 {K0,K1}; lanes16-31 {K2,K3}
//   B 4x16 f32: lanes0-15 n=0..15 {K0,K1}; lanes16-31 {K2,K3}
//   C/D: vgpr i, lane L -> (m = i + 8*(L>>4), n = L&15)
// ===========================================================================
__global__ void k_gemm_wmma(const float* __restrict__ A, int lda, int transA,
                            const float* __restrict__ B, int ldb, int transB,
                            const float* __restrict__ Csrc, int ldcs,
                            float* __restrict__ Cdst, int ldcd,
                            int K, float scale)
{
    __shared__ float sA[16][68];
    __shared__ float sB[16][68];
    const int lane = threadIdx.x;
    const int m0 = blockIdx.y * 16, n0 = blockIdx.x * 16;
    const int mloc = lane & 15, kh = lane >> 4;
    v8f acc = {};
    for (int kk = 0; kk < K; kk += 64) {
        for (int t = lane; t < 16 * 64; t += 32) {
            int m = t >> 6, k = t & 63;
            sA[m][k] = transA ? A[(size_t)(kk + k) * lda + (m0 + m)]
                              : A[(size_t)(m0 + m) * lda + (kk + k)];
        }
        for (int t = lane; t < 16 * 64; t += 32) {
            int n = t >> 6, k = t & 63;
            sB[n][k] = transB ? B[(size_t)(n0 + n) * ldb + (kk + k)]
                              : B[(size_t)(kk + k) * ldb + (n0 + n)];
        }
        __syncthreads();
#if HAVE_WMMA_F32
        #pragma unroll
        for (int p = 0; p < 16; ++p) {
            v2f a, b;
            a.x = sA[mloc][p * 4 + 2 * kh];
            a.y = sA[mloc][p * 4 + 2 * kh + 1];
            b.x = sB[mloc][p * 4 + 2 * kh];
            b.y = sB[mloc][p * 4 + 2 * kh + 1];
            acc = __builtin_amdgcn_wmma_f32_16x16x4_f32(
                false, a, false, b, (short)0, acc, false, false);
        }
#else
        for (int i = 0; i < 8; ++i) {
            int mr = i + 8 * kh;
            float s = 0.f;
            for (int k = 0; k < 64; ++k) s += sA[mr][k] * sB[mloc][k];
            acc[i] += s;
        }
#endif
        __syncthreads();
    }
    #pragma unroll
    for (int i = 0; i < 8; ++i) {
        int r = m0 + i + 8 * kh, c = n0 + mloc;
        float v = scale * acc[i];
        if (Csrc) v += Csrc[(size_t)r * ldcs + c];
        Cdst[(size_t)r * ldcd + c] = v;
    }
}

// ---------------- build K_e = var*exp(-0.5*||(x_r-x_c)/l||^2) + nv*I -------
__global__ void k_build_K(const float* __restrict__ X,
                          const float* __restrict__ leng,  // &lens[e*DIN_]
                          const float* __restrict__ var,
                          const float* __restrict__ nv,
                          float* __restrict__ Kout)
{
    int c = blockIdx.x * 16 + threadIdx.x;
    int r = blockIdx.y * 16 + threadIdx.y;
    float il[DIN_];
    #pragma unroll
    for (int d = 0; d < DIN_; ++d) il[d] = 1.0f / leng[d];
    float d2 = 0.f;
    #pragma unroll
    for (int d = 0; d < DIN_; ++d) {
        float t = (X[(size_t)r * DIN_ + d] - X[(size_t)c * DIN_ + d]) * il[d];
        d2 += t * t;
    }
    float v = var[0] * __expf(-0.5f * d2);
    if (r == c) v += nv[0];
    Kout[(size_t)r * N_ + c] = v;
}

// ---------------- factor 64x64 diagonal block, write L and inv(L) ----------
__global__ void k_chol_diag(float* __restrict__ A, float* __restrict__ invD, int kb)
{
    __shared__ float L[BS_][BS_ + 1];
    __shared__ float V[BS_][BS_ + 1];
    int tid = threadIdx.x;                    // 64 threads
    for (int r = 0; r < BS_; ++r)
        L[r][tid] = A[(size_t)(kb * BS_ + r) * N_ + kb * BS_ + tid];
    __syncthreads();
    for (int j = 0; j < BS_; ++j) {
        if (tid == 0) L[j][j] = sqrtf(fmaxf(L[j][j], 1e-30f));
        __syncthreads();
        float dinv = 1.0f / L[j][j];
        if (tid > j) L[tid][j] *= dinv;
        __syncthreads();
        if (tid > j) {
            float f = L[tid][j];
            for (int c = j + 1; c <= tid; ++c) L[tid][c] -= f * L[c][j];
        }
        __syncthreads();
    }
    // invert lower-triangular L: thread tid computes column tid
    for (int i = 0; i < BS_; ++i) {
        float s = (i == tid) ? 1.0f : 0.0f;
        for (int p = tid; p < i; ++p) s -= L[i][p] * V[p][tid];
        V[i][tid] = (i >= tid) ? s / L[i][i] : 0.0f;
    }
    __syncthreads();
    for (int r = 0; r < BS_; ++r) {
        A[(size_t)(kb * BS_ + r) * N_ + kb * BS_ + tid] = (tid <= r) ? L[r][tid] : 0.0f;
        invD[r * BS_ + tid] = V[r][tid];
    }
}

// ---------------- strided 2D copy --------------------------------------
__global__ void k_copy(const float* __restrict__ src, int lds_,
                       float* __restrict__ dst, int ldd, int rows, int cols)
{
    int t = blockIdx.x * 256 + threadIdx.x;
    if (t < rows * cols) {
        int r = t / cols, c = t - r * cols;
        dst[(size_t)r * ldd + c] = src[(size_t)r * lds_ + c];
    }
}

// ---------------- beta = iK @ y_e --------------------------------------
__global__ void k_beta(const float* __restrict__ iK, const float* __restrict__ Y,
                       float* __restrict__ beta, int e)
{
    int r = blockIdx.x * 256 + threadIdx.x;
    if (r < N_) {
        float s = 0.f;
        for (int c = 0; c < N_; ++c) s += iK[(size_t)r * N_ + c] * Y[(size_t)c * E_ + e];
        beta[r] = s;
    }
}

// ---------------- inp/scaled/logk precompute ---------------------------
__global__ void k_prep(const float* __restrict__ X, const float* __restrict__ xk,
                       const float* __restrict__ u, const float* __restrict__ lens,
                       const float* __restrict__ vars,
                       float* __restrict__ inp, float* __restrict__ scaled,
                       float* __restrict__ logk)
{
    int n = blockIdx.x * 256 + threadIdx.x;
    if (n >= N_) return;
    float ip[DIN_];
    #pragma unroll
    for (int d = 0; d < DIN_; ++d)
        ip[d] = X[(size_t)n * DIN_ + d] - (d < D_ ? xk[d] : u[d - D_]);
    #pragma unroll
    for (int d = 0; d < DP_; ++d) inp[(size_t)n * DP_ + d] = (d < DIN_) ? ip[d] : 0.f;
    for (int e = 0; e < E_; ++e) {
        float acc = 0.f;
        float* sc = scaled + ((size_t)e * N_ + n) * DP_;
        #pragma unroll
        for (int d = 0; d < DIN_; ++d) {
            float le = lens[e * DIN_ + d];
            float nn = ip[d] / le;
            sc[d] = nn / le;            // inp * ilen2
            acc += nn * nn;
        }
        sc[10] = 0.f; sc[11] = 0.f;
        logk[(size_t)e * N_ + n] = logf(vars[e]) - 0.5f * acc;
    }
}

__global__ void k_xvar(const float* __restrict__ Pk, float* __restrict__ XV)
{
    int t = threadIdx.x;
    if (t < DIN_ * DIN_) {
        int a = t / DIN_, b = t - a * DIN_;
        XV[t] = (a < D_ && b < D_) ? Pk[a * D_ + b] : 0.0f;
    }
}

// ---------------- per-pair small solve: iRS = 0.5*R^{-1}Xvar, 1/sqrt(detR)
__global__ void k_pair_setup(const float* __restrict__ XV, const float* __restrict__ lens,
                             int ei, int ej, float* __restrict__ iRS,
                             float* __restrict__ isdR)
{
    float R[DIN_][DIN_], G[DIN_][DIN_];
    float s2[DIN_];
    for (int d = 0; d < DIN_; ++d) {
        float a = lens[ei * DIN_ + d], b = lens[ej * DIN_ + d];
        s2[d] = 1.f / (a * a) + 1.f / (b * b);
    }
    for (int a = 0; a < DIN_; ++a)
        for (int b = 0; b < DIN_; ++b) {
            R[a][b] = XV[a * DIN_ + b] * s2[b] + (a == b ? 1.f : 0.f);
            G[a][b] = XV[a * DIN_ + b];
        }
    float det = 1.f;
    for (int p = 0; p < DIN_; ++p) {
        int piv = p; float mx = fabsf(R[p][p]);
        for (int r = p + 1; r < DIN_; ++r)
            if (fabsf(R[r][p]) > mx) { mx = fabsf(R[r][p]); piv = r; }
        if (piv != p) {
            for (int c = 0; c < DIN_; ++c) {
                float t = R[p][c]; R[p][c] = R[piv][c]; R[piv][c] = t;
                t = G[p][c]; G[p][c] = G[piv][c]; G[piv][c] = t;
            }
            det = -det;
        }
        float d0 = R[p][p]; det *= d0; float iv = 1.f / d0;
        for (int c = 0; c < DIN_; ++c) { R[p][c] *= iv; G[p][c] *= iv; }
        for (int r = 0; r < DIN_; ++r) if (r != p) {
            float f = R[r][p];
            for (int c = 0; c < DIN_; ++c) { R[r][c] -= f * R[p][c]; G[r][c] -= f * G[p][c]; }
        }
    }
    for (int a = 0; a < DIN_; ++a)
        for (int b = 0; b < DIN_; ++b) iRS[a * DIN_ + b] = 0.5f * G[a][b];
    *isdR = rsqrtf(det);
}

// U = scaled_i @ iRS, U[.,10] = rowdot(U, scaled_i), U[.,11]=0
__global__ void k_project(const float* __restrict__ sc, const float* __restrict__ iRS,
                          float* __restrict__ U)
{
    int n = blockIdx.x * 256 + threadIdx.x;
    if (n >= N_) return;
    const float* s = sc + (size_t)n * DP_;
    float uv[DIN_]; float ai = 0.f;
    #pragma unroll
    for (int d = 0; d < DIN_; ++d) {
        float t = 0.f;
        #pragma unroll
        for (int a = 0; a < DIN_; ++a) t += s[a] * iRS[a * DIN_ + d];
        uv[d] = t; ai += t * s[d];
    }
    float* o = U + (size_t)n * DP_;
    #pragma unroll
    for (int d = 0; d < DIN_; ++d) o[d] = uv[d];
    o[10] = ai; o[11] = 0.f;
}

// ---------------- fused pair term: accumulates b_i Qm b_j and iK.*Qm ----
// one wave per 64x64 tile; cross term via f32 WMMA with K=12 (padded)
__global__ void k_pair_main(const float* __restrict__ U,    // N x 12, [10]=ai
                            const float* __restrict__ Sj,   // scaled_j, N x 12 (10,11=0)
                            const float* __restrict__ Wn,   // N x 12, [10]=aj
                            const float* __restrict__ lki, const float* __restrict__ lkj,
                            const float* __restrict__ bi, const float* __restrict__ bj,
                            const float* __restrict__ iK,   // nullable
                            float* __restrict__ partial)
{
    __shared__ float sU[64][13];
    __shared__ float sS[64][13];
    __shared__ float svn[64][3];    // bi, logki, ai
    __shared__ float svm[64][3];    // bj, logkj, aj
    __shared__ float red[32];
    int lane = threadIdx.x;
    int n0 = blockIdx.y * 64, m0 = blockIdx.x * 64;
    for (int t = lane; t < 64 * DP_; t += 32) {
        int r = t / DP_, c = t - r * DP_;
        sU[r][c] = U[(size_t)(n0 + r) * DP_ + c];
        sS[r][c] = Sj[(size_t)(m0 + r) * DP_ + c];
    }
    for (int t = lane; t < 64; t += 32) {
        svn[t][0] = bi[n0 + t]; svn[t][1] = lki[n0 + t];
        svn[t][2] = U[(size_t)(n0 + t) * DP_ + 10];
        svm[t][0] = bj[m0 + t]; svm[t][1] = lkj[m0 + t];
        svm[t][2] = Wn[(size_t)(m0 + t) * DP_ + 10];
    }
    __syncthreads();
    int mloc = lane & 15, kh = lane >> 4;
    float acc1 = 0.f, acc2 = 0.f;
    for (int tn = 0; tn < 4; ++tn)
    for (int tm = 0; tm < 4; ++tm) {
        v8f c = {};
#if HAVE_WMMA_F32
        #pragma unroll
        for (int p = 0; p < 3; ++p) {        // K = 12 (cols 10,11 of Sj are zero)
            v2f a, b;
            a.x = sU[tn * 16 + mloc][p * 4 + 2 * kh];
            a.y = sU[tn * 16 + mloc][p * 4 + 2 * kh + 1];
            b.x = sS[tm * 16 + mloc][p * 4 + 2 * kh];
            b.y = sS[tm * 16 + mloc][p * 4 + 2 * kh + 1];
            c = __builtin_amdgcn_wmma_f32_16x16x4_f32(
                false, a, false, b, (short)0, c, false, false);
        }
#else
        for (int i = 0; i < 8; ++i) {
            int nr = tn * 16 + i + 8 * kh;
            float s = 0.f;
            for (int d = 0; d < DIN_; ++d) s += sU[nr][d] * sS[tm * 16 + mloc][d];
            c[i] = s;
        }
#endif
        #pragma unroll
        for (int i = 0; i < 8; ++i) {
            int nl = tn * 16 + i + 8 * kh;
            int ml = tm * 16 + mloc;
            float q = __expf(svn[nl][1] + svm[ml][1] + svn[nl][2] + svm[ml][2] + 2.f * c[i]);
            acc1 += svn[nl][0] * svm[ml][0] * q;
            if (iK) acc2 += iK[(size_t)(n0 + nl) * N_ + (m0 + ml)] * q;
        }
    }
    red[lane] = acc1; __syncthreads();
    for (int s = 16; s > 0; s >>= 1) { if (lane < s) red[lane] += red[lane + s]; __syncthreads(); }
    float r1 = red[0]; __syncthreads();
    red[lane] = acc2; __syncthreads();
    for (int s = 16; s > 0; s >>= 1) { if (lane < s) red[lane] += red[lane + s]; __syncthreads(); }
    if (lane == 0) {
        int b = (blockIdx.y * gridDim.x + blockIdx.x) * 2;
        partial[b] = r1; partial[b + 1] = red[0];
    }
}

__global__ void k_pair_reduce(const float* __restrict__ partial, float* __restrict__ out2)
{
    __shared__ float red[256];
    int tid = threadIdx.x;
    float a = 0.f, b = 0.f;
    for (int i = tid; i < 1024; i += 256) { a += partial[i * 2]; b += partial[i * 2 + 1]; }
    red[tid] = a; __syncthreads();
    for (int s = 128; s > 0; s >>= 1) { if (tid < s) red[tid] += red[tid + s]; __syncthreads(); }
    if (tid == 0) out2[0] = red[0];
    __syncthreads();
    red[tid] = b; __syncthreads();
    for (int s = 128; s > 0; s >>= 1) { if (tid < s) red[tid] += red[tid + s]; __syncthreads(); }
    if (tid == 0) out2[1] = red[0];
}

// ---------------- mean/cov small setup: G = iQ/ll, H = G@Xvar, c -------
__global__ void k_mean_setup(const float* __restrict__ XV, const float* __restrict__ leng,
                             const float* __restrict__ var, float* __restrict__ mb)
{
    float B[DIN_][DIN_], Gi[DIN_][DIN_];
    for (int a = 0; a < DIN_; ++a)
        for (int b = 0; b < DIN_; ++b) {
            B[a][b] = XV[a * DIN_ + b] / (leng[a] * leng[b]) + (a == b ? 1.f : 0.f);
            Gi[a][b] = (a == b) ? 1.f : 0.f;
        }
    float det = 1.f;
    for (int p = 0; p < DIN_; ++p) {
        int piv = p; float mx = fabsf(B[p][p]);
        for (int r = p + 1; r < DIN_; ++r)
            if (fabsf(B[r][p]) > mx) { mx = fabsf(B[r][p]); piv = r; }
        if (piv != p) {
            for (int c = 0; c < DIN_; ++c) {
                float t = B[p][c]; B[p][c] = B[piv][c]; B[piv][c] = t;
                t = Gi[p][c]; Gi[p][c] = Gi[piv][c]; Gi[piv][c] = t;
            }
            det = -det;
        }
        float d0 = B[p][p]; det *= d0; float iv = 1.f / d0;
        for (int c = 0; c < DIN_; ++c) { B[p][c] *= iv; Gi[p][c] *= iv; }
        for (int r = 0; r < DIN_; ++r) if (r != p) {
            float f = B[r][p];
            for (int c = 0; c < DIN_; ++c) { B[r][c] -= f * B[p][c]; Gi[r][c] -= f * Gi[p][c]; }
        }
    }
    // G = iQ / (l_a l_b); H = G @ Xvar
    for (int a = 0; a < DIN_; ++a)
        for (int b = 0; b < DIN_; ++b)
            mb[a * DIN_ + b] = Gi[a][b] / (leng[a] * leng[b]);
    for (int a = 0; a < DIN_; ++a)
        for (int b = 0; b < DIN_; ++b) {
            float s = 0.f;
            for (int p = 0; p < DIN_; ++p) s += mb[a * DIN_ + p] * XV[p * DIN_ + b];
            mb[100 + a * DIN_ + b] = s;
        }
    mb[200] = var[0] * rsqrtf(det);
}

__global__ void k_mean_reduce(const float* __restrict__ inp, const float* __restrict__ beta,
                              const float* __restrict__ mb,
                              float* __restrict__ Mout, float* __restrict__ Vout)
{
    __shared__ float sG[DIN_ * DIN_];
    __shared__ float red[256];
    __shared__ float res[DIN_ + 1];
    int tid = threadIdx.x;
    for (int t = tid; t < DIN_ * DIN_; t += 256) sG[t] = mb[t];
    __syncthreads();
    float s0 = 0.f, tacc[DIN_];
    #pragma unroll
    for (int d = 0; d < DIN_; ++d) tacc[d] = 0.f;
    for (int n = tid; n < N_; n += 256) {
        const float* ip = inp + (size_t)n * DP_;
        float q = 0.f;
        #pragma unroll
        for (int a = 0; a < DIN_; ++a) {
            float y = 0.f;
            #pragma unroll
            for (int b = 0; b < DIN_; ++b) y += sG[a * DIN_ + b] * ip[b];
            q += ip[a] * y;
        }
        float lb = __expf(-0.5f * q) * beta[n];
        s0 += lb;
        #pragma unroll
        for (int d = 0; d < DIN_; ++d) tacc[d] += lb * ip[d];
    }
    for (int ch = 0; ch <= DIN_; ++ch) {
        red[tid] = (ch == 0) ? s0 : tacc[ch - 1];
        __syncthreads();
        for (int s = 128; s > 0; s >>= 1) { if (tid < s) red[tid] += red[tid + s]; __syncthreads(); }
        if (tid == 0) res[ch] = red[0];
        __syncthreads();
    }
    if (tid == 0) {
        float c = mb[200];
        Mout[0] = c * res[0];
        for (int d = 0; d < DIN_; ++d) {
            float v = 0.f;
            for (int a = 0; a < DIN_; ++a) v += res[1 + a] * mb[100 + a * DIN_ + d];
            Vout[d] = c * v;
        }
    }
}

// ---------------- final assembly: out = [M(6) | S+Qn(36) | V[:D](36)] --
__global__ void k_finalize(const float* __restrict__ Mb, const float* __restrict__ Vc,
                           const float* __restrict__ sacc, const float* __restrict__ isd,
                           const float* __restrict__ vars, const float* __restrict__ Qn,
                           float* __restrict__ out)
{
    int tid = threadIdx.x;
    if (tid < E_) out[tid] = Mb[tid];
    if (tid < E_ * E_) {
        int i = tid / E_, j = tid - i * E_;
        float v = (sacc[tid * 2] - sacc[tid * 2 + 1]) * isd[tid];
        if (i == j) v += vars[i];
        v -= Mb[i] * Mb[j];
        v += Qn[i * D_ + j];
        out[E_ + tid] = v;
        int d = i, e = j;                       // V[:D] is D x E
        out[E_ + E_ * E_ + tid] = Vc[e * DP_ + d];
    }
}

// ===========================================================================
static void gemm(hipStream_t st, const float* A, int lda, int tA,
                 const float* B, int ldb, int tB,
                 const float* Cs, int ldcs, float* Cd, int ldcd,
                 int K, float scale, int M, int Nc)
{
    dim3 g(Nc / 16, M / 16);
    k_gemm_wmma<<<g, dim3(32), 0, st>>>(A, lda, tA, B, ldb, tB, Cs, ldcs, Cd, ldcd, K, scale);
}

extern "C" void kernel_launch(void* const* d_in, const int* in_sizes, int n_in,
                              void* d_out, int out_size, void* d_ws, size_t ws_size,
                              hipStream_t stream)
{
    (void)in_sizes; (void)n_in; (void)out_size; (void)ws_size;
    const float* X    = (const float*)d_in[0];
    const float* Y    = (const float*)d_in[1];
    const float* xk   = (const float*)d_in[2];
    const float* Pk   = (const float*)d_in[3];
    const float* u    = (const float*)d_in[4];
    const float* Qn   = (const float*)d_in[5];
    const float* vars = (const float*)d_in[6];
    const float* lens = (const float*)d_in[7];
    const float* nois = (const float*)d_in[8];
    float* out = (float*)d_out;

    const size_t NN = (size_t)N_ * N_;
    float* w = (float*)d_ws;
    size_t o = 0;
    float* A    = w + o; o += NN;
    float* Z    = w + o; o += NN;
    float* IK   = w + o; o += NN;
    float* TB   = w + o; o += (size_t)N_ * BS_;
    float* INVD = w + o; o += (size_t)NB_ * BS_ * BS_;
    float* INP  = w + o; o += (size_t)N_ * DP_;
    float* SC   = w + o; o += (size_t)E_ * N_ * DP_;
    float* LOGK = w + o; o += (size_t)E_ * N_;
    float* BETA = w + o; o += (size_t)E_ * N_;
    float* UB   = w + o; o += (size_t)N_ * DP_;
    float* WB   = w + o; o += (size_t)N_ * DP_;
    float* XV   = w + o; o += 128;
    float* IRS  = w + o; o += 128;
    float* ISD  = w + o; o += 64;
    float* PART = w + o; o += 2048;
    float* SACC = w + o; o += 128;
    float* MNB  = w + o; o += 256;
    float* MB   = w + o; o += 16;
    float* VC   = w + o; o += 128;

    k_xvar<<<1, 128, 0, stream>>>(Pk, XV);
    k_prep<<<(N_ + 255) / 256, 256, 0, stream>>>(X, xk, u, lens, vars, INP, SC, LOGK);

    // ---------- Phase A: per latent, K-inverse via WMMA-f32 blocked chol ----
    for (int e = 0; e < E_; ++e) {
        k_build_K<<<dim3(N_ / 16, N_ / 16), dim3(16, 16), 0, stream>>>(
            X, lens + e * DIN_, vars + e, nois + e, A);

        // blocked right-looking Cholesky
        for (int k = 0; k < NB_; ++k) {
            k_chol_diag<<<1, BS_, 0, stream>>>(A, INVD + (size_t)k * BS_ * BS_, k);
            int rows = N_ - (k + 1) * BS_;
            if (rows > 0) {
                float* P = A + (size_t)(k + 1) * BS_ * N_ + k * BS_;
                // panel TRSM:  P <- P @ inv(L_kk)^T   (via temp to avoid in-place race)
                gemm(stream, P, N_, 0, INVD + (size_t)k * BS_ * BS_, BS_, 1,
                     nullptr, 0, TB, BS_, BS_, 1.0f, rows, BS_);
                k_copy<<<(rows * BS_ + 255) / 256, 256, 0, stream>>>(TB, BS_, P, N_, rows, BS_);
                // trailing SYRK: C <- C - P P^T (full square; symmetric)
                float* C = A + (size_t)(k + 1) * BS_ * N_ + (k + 1) * BS_;
                gemm(stream, P, N_, 0, P, N_, 1, C, N_, C, N_, BS_, -1.0f, rows, rows);
            }
        }

        // forward: Z = L^{-1}   (lower triangular, Z zero-initialized)
        hipMemsetAsync(Z, 0, NN * sizeof(float), stream);
        k_copy<<<(BS_ * BS_ + 255) / 256, 256, 0, stream>>>(INVD, BS_, Z, N_, BS_, BS_);
        for (int i = 1; i < NB_; ++i) {
            gemm(stream, A + (size_t)i * BS_ * N_, N_, 0, Z, N_, 0,
                 nullptr, 0, TB, N_, i * BS_, -1.0f, BS_, i * BS_);
            gemm(stream, INVD + (size_t)i * BS_ * BS_, BS_, 0, TB, N_, 0,
                 nullptr, 0, Z + (size_t)i * BS_ * N_, N_, BS_, 1.0f, BS_, i * BS_);
            k_copy<<<(BS_ * BS_ + 255) / 256, 256, 0, stream>>>(
                INVD + (size_t)i * BS_ * BS_, BS_,
                Z + (size_t)i * BS_ * N_ + i * BS_, N_, BS_, BS_);
        }

        // backward: iK = L^{-T} Z  (row blocks bottom-up)
        gemm(stream, INVD + (size_t)(NB_ - 1) * BS_ * BS_, BS_, 1,
             Z + (size_t)(NB_ - 1) * BS_ * N_, N_, 0,
             nullptr, 0, IK + (size_t)(NB_ - 1) * BS_ * N_, N_, BS_, 1.0f, BS_, N_);
        for (int i = NB_ - 2; i >= 0; --i) {
            int Kb = (NB_ - 1 - i) * BS_;
            gemm(stream, A + (size_t)(i + 1) * BS_ * N_ + i * BS_, N_, 1,
                 IK + (size_t)(i + 1) * BS_ * N_, N_, 0,
                 Z + (size_t)i * BS_ * N_, N_, TB, N_, Kb, -1.0f, BS_, N_);
            gemm(stream, INVD + (size_t)i * BS_ * BS_, BS_, 1, TB, N_, 0,
                 nullptr, 0, IK + (size_t)i * BS_ * N_, N_, BS_, 1.0f, BS_, N_);
        }

        k_beta<<<(N_ + 255) / 256, 256, 0, stream>>>(IK, Y, BETA + (size_t)e * N_, e);

        // diagonal pair term (e,e) while iK is live
        k_pair_setup<<<1, 1, 0, stream>>>(XV, lens, e, e, IRS, ISD + e * E_ + e);
        k_project<<<(N_ + 255) / 256, 256, 0, stream>>>(SC + (size_t)e * N_ * DP_, IRS, UB);
        k_pair_main<<<dim3(N_ / 64, N_ / 64), 32, 0, stream>>>(
            UB, SC + (size_t)e * N_ * DP_, UB,
            LOGK + (size_t)e * N_, LOGK + (size_t)e * N_,
            BETA + (size_t)e * N_, BETA + (size_t)e * N_, IK, PART);
        k_pair_reduce<<<1, 256, 0, stream>>>(PART, SACC + (e * E_ + e) * 2);
    }

    // ---------- Phase B: off-diagonal pair terms (no iK needed) ------------
    for (int i = 0; i < E_; ++i)
        for (int j = 0; j < E_; ++j) {
            if (i == j) continue;
            k_pair_setup<<<1, 1, 0, stream>>>(XV, lens, i, j, IRS, ISD + i * E_ + j);
            k_project<<<(N_ + 255) / 256, 256, 0, stream>>>(SC + (size_t)i * N_ * DP_, IRS, UB);
            k_project<<<(N_ + 255) / 256, 256, 0, stream>>>(SC + (size_t)j * N_ * DP_, IRS, WB);
            k_pair_main<<<dim3(N_ / 64, N_ / 64), 32, 0, stream>>>(
                UB, SC + (size_t)j * N_ * DP_, WB,
                LOGK + (size_t)i * N_, LOGK + (size_t)j * N_,
                BETA + (size_t)i * N_, BETA + (size_t)j * N_, nullptr, PART);
            k_pair_reduce<<<1, 256, 0, stream>>>(PART, SACC + (i * E_ + j) * 2);
        }

    // ---------- Phase C: mean / input-output covariance / assembly ---------
    for (int e = 0; e < E_; ++e) {
        k_mean_setup<<<1, 1, 0, stream>>>(XV, lens + e * DIN_, vars + e, MNB);
        k_mean_reduce<<<1, 256, 0, stream>>>(INP, BETA + (size_t)e * N_, MNB,
                                             MB + e, VC + e * DP_);
    }
    k_finalize<<<1, 64, 0, stream>>>(MB, VC, SACC, ISD, vars, Qn, out);
}